// chord_89120571392668
// MI455X (gfx1250) — compile-verified
//
#include <hip/hip_runtime.h>
#include <stdint.h>

// ---------------------------------------------------------------------------
// Memory-bound chord-duration kernel for MI455X (gfx1250).
//   - TDM (tensor_load_to_lds, th:TH_LOAD_NT) DMAs each wave's contiguous
//     sample slab into LDS, double-issued so compute overlaps DMA
//     (TENSORcnt completes in-order per wave).
//   - wave32 ballots build the 16x16 "hold" matrix as bitmasks; run-length =
//     ctz of the inverted selected mask (== the reference's cumprod-sum).
//   - ballot[idx] picked by a binary cndmask tree over idx bits.
//   - ds_load_b64 consumption is LDS-bank-conflict-free (stride 26 dwords).
//   - WEIGHTS[idx] selected via 3-bit packed immediate (exact in f32).
// ---------------------------------------------------------------------------

typedef __attribute__((ext_vector_type(4))) unsigned int u32x4;
typedef __attribute__((ext_vector_type(8))) unsigned int u32x8;

// 1-D TDM load: elems8 elements of 8 bytes, global -> LDS byte offset.
// D# layout per CDNA5 ISA ch.8 (group0 128b, group1 256b; groups 2/3 = NULL).
__device__ __forceinline__ void tdm_load_1d(unsigned int lds_byte_off,
                                            const void* gsrc,
                                            unsigned int elems8) {
  uint64_t ga = (uint64_t)(uintptr_t)gsrc;
  u32x4 g0;
  g0.s0 = 1u;                                              // count=1 (valid user D#)
  g0.s1 = lds_byte_off;                                    // lds_addr (bytes)
  g0.s2 = (unsigned int)ga;                                // global_addr[31:0]
  g0.s3 = (unsigned int)((ga >> 32) & 0x01FFFFFFull)       // global_addr[56:32]
        | 0x80000000u;                                     // type=2 ("image")
  u32x8 g1;
  g1.s0 = 3u << 16;                                        // wg_mask=0, data_size=8B
  g1.s1 = (elems8 & 0xFFFFu) << 16;                        // tensor_dim0[15:0]
  g1.s2 = ((elems8 >> 16) & 0xFFFFu) | (1u << 16);         // tensor_dim0[31:16] | tensor_dim1=1
  g1.s3 = (elems8 & 0xFFFFu) << 16;                        // tile_dim0 (16-bit)
  g1.s4 = 0u;                                              // tile_dim1=0, tile_dim2=0 (unused)
  g1.s5 = elems8;                                          // tensor_dim0_stride[31:0]
  g1.s6 = 0u;                                              // stride hi / dim1_stride lo
  g1.s7 = 0u;                                              // dim1_stride hi
  // Non-temporal: 218MB read-once stream must not displace 192MB L2.
  asm volatile("tensor_load_to_lds %0, %1 th:TH_LOAD_NT" :: "s"(g0), "s"(g1) : "memory");
}

#define SAMPLE_FLOATS 416   // 16 timesteps * 26 features
#define SAMPLE_BYTES  1664
#define SPW           8     // samples per wave
#define SPB           64    // samples per block (8 waves)

__global__ void __launch_bounds__(256)
chord_kernel(const float* __restrict__ data, const float* __restrict__ label,
             float* __restrict__ out, int B) {
  extern __shared__ float smem[];
  const int lane = threadIdx.x & 31;
  const int wid  = threadIdx.x >> 5;
  const int g    = lane >> 4;       // which of the 2 samples this wave-pass
  const int t    = lane & 15;       // timestep owned by this lane

  const int blockBase = blockIdx.x * SPB;
  const int waveBase  = blockBase + wid * SPW;
  const int remain    = B - waveBase;
  const bool full     = remain >= SPW;

  // LDS byte offset of this wave's slab (flat-shared low 32 bits == LDS offset)
  const unsigned int ldsOff =
      (unsigned int)(uintptr_t)smem + (unsigned int)(wid * SAMPLE_BYTES * SPW);

  if (full) {
    const float* src = data + (size_t)waveBase * SAMPLE_FLOATS;
    tdm_load_1d(ldsOff,                     src,                         832u); // samples 0..3
    tdm_load_1d(ldsOff + 4 * SAMPLE_BYTES,  src + 4 * SAMPLE_FLOATS,     832u); // samples 4..7
  } else if (remain > 0) {
    tdm_load_1d(ldsOff, data + (size_t)waveBase * SAMPLE_FLOATS,
                (unsigned int)remain * (SAMPLE_BYTES / 8));
  }

  if (full) { asm volatile("s_wait_tensorcnt 0x1" ::: "memory"); }  // first DMA done
  else      { asm volatile("s_wait_tensorcnt 0x0" ::: "memory"); }

  // WEIGHTS packed as multiples of 0.2f in 3-bit fields (idx 0..11):
  // {0.6,0,0.2,0,1.0,0.2,0,0.6,0,0.2,0,0.2} -> {3,0,1,0,5,1,0,3,0,1,0,1}
  // 3*0.2f and 5*0.2f round to exactly 0.6f and 1.0f in f32.
  const uint64_t WPACK = (3ull << 0) | (1ull << 6) | (5ull << 12) | (1ull << 15)
                       | (3ull << 21) | (1ull << 27) | (1ull << 33);

  #pragma unroll
  for (int p = 0; p < 4; ++p) {            // 4 passes x 2 samples = 8 samples/wave
    if (p == 2) { asm volatile("s_wait_tensorcnt 0x0" ::: "memory"); } // second DMA

    const int sl = wid * SPW + p * 2 + g;  // sample index within block
    const int s  = blockBase + sl;         // global sample index

    // Lane t reads its timestep's 26 floats: 13x ds_load_b64, 8B-aligned,
    // bank-conflict-free across all 32 lanes.
    const float2* sp2 =
        (const float2*)(smem + sl * SAMPLE_FLOATS + t * 26);
    float v[26];
    #pragma unroll
    for (int k = 0; k < 13; ++k) { float2 x = sp2[k]; v[2*k] = x.x; v[2*k+1] = x.y; }

    // max / argmax over 12 pitch columns (first-max tie rule like jnp.argmax)
    float mx = v[0]; int idx = 0;
    #pragma unroll
    for (int c = 1; c < 12; ++c) { if (v[c] > mx) { mx = v[c]; idx = c; } }

    // Wave-wide M matrix: ballot bit j == (hold[j][c] > 0.7). Lanes 0-15 are
    // sample A's timesteps, 16-31 sample B's. Each lane needs ballot[idx].
    unsigned int b[12];
    #pragma unroll
    for (int c = 0; c < 12; ++c)
      b[c] = __builtin_amdgcn_ballot_w32(v[13 + c] > 0.7f); // hold col c = data col 13+c

    // Binary cndmask tree over idx bits: 11 selects, 4 lane-mask compares.
    const bool o1 = (idx & 1) != 0;
    const unsigned int m01 = o1 ? b[1]  : b[0];
    const unsigned int m23 = o1 ? b[3]  : b[2];
    const unsigned int m45 = o1 ? b[5]  : b[4];
    const unsigned int m67 = o1 ? b[7]  : b[6];
    const unsigned int m89 = o1 ? b[9]  : b[8];
    const unsigned int mAB = o1 ? b[11] : b[10];
    const bool o2 = (idx & 2) != 0;
    const unsigned int m03 = o2 ? m23 : m01;
    const unsigned int m47 = o2 ? m67 : m45;
    const unsigned int m8B = o2 ? mAB : m89;
    const int hi = idx >> 2;                 // 0, 1 or 2
    unsigned int mask = (hi == 1) ? m47 : m03;
    mask = (hi == 2) ? m8B : mask;

    // WEIGHTS[idx] via packed shift (replaces a 12-deep cndmask chain)
    const float wsel = 0.2f * (float)((unsigned int)(WPACK >> (3 * idx)) & 7u);

    // run = # consecutive set bits of M[t, t+1..15]  (== cumprod-sum)
    const unsigned int grp  = (mask >> (g * 16)) & 0xFFFFu;
    const unsigned int bits = grp >> (t + 1);       // bit k <-> j = t+1+k
    const unsigned int run  = (unsigned int)__builtin_ctz(~bits); // auto-capped at 15-t

    float cons = (mx >= 0.7f) ? mx * wsel * (1.0f + (float)run) : 0.0f;

    // sum over the 16 timesteps of this sample (xor masks 1,2,4,8 stay in-group)
    #pragma unroll
    for (int off = 1; off < 16; off <<= 1)
      cons += __shfl_xor(cons, off, 32);

    if (t == 0 && s < B) out[s] = cons / label[s * 15 + 14];
  }
}

extern "C" void kernel_launch(void* const* d_in, const int* in_sizes, int n_in,
                              void* d_out, int out_size, void* d_ws, size_t ws_size,
                              hipStream_t stream) {
  (void)n_in; (void)out_size; (void)d_ws; (void)ws_size;
  const float* data  = (const float*)d_in[0];
  const float* label = (const float*)d_in[1];
  float* out         = (float*)d_out;
  const int B      = in_sizes[0] / SAMPLE_FLOATS;   // 131072
  const int blocks = (B + SPB - 1) / SPB;           // 2048
  chord_kernel<<<blocks, 256, SPB * SAMPLE_BYTES, stream>>>(data, label, out, B);
}